// GNN_63007170232901
// MI455X (gfx1250) — compile-verified
//
#include <hip/hip_runtime.h>
#include <hip/hip_bf16.h>

typedef __attribute__((ext_vector_type(2))) float v2f;
typedef __attribute__((ext_vector_type(8))) float v8f;

#define IN_DIM  128
#define HID     256
#define NCLASS  10
#define NGRAPH  512
#define BN_EPS  1e-5f

// ---------------------------------------------------------------------------
// Zero-fill (float4 granularity; all sizes used are multiples of 4 floats)
// ---------------------------------------------------------------------------
__global__ __launch_bounds__(256) void k_zero(float4* __restrict__ p, long n4) {
    long i = (long)blockIdx.x * blockDim.x + threadIdx.x;
    if (i < n4) p[i] = make_float4(0.f, 0.f, 0.f, 0.f);
}

// ---------------------------------------------------------------------------
// FP32 WMMA GEMM:  C[M,N] = A[M,K] @ B[K,N] + bias  (optional fused ReLU)
//
// Grid: (ceil(M/32/8), N/32).  All 8 waves of a block share one 32-column
// B panel (tn = blockIdx.y); the block stages the K x 32 panel into LDS
// once, pair-swizzled as Bs[k/2][n][k%2] so the (k,k+1) pair a lane needs
// for its fragment is one DS_LOAD_B64.  Each wave then owns a 32x32 C tile
// (2x2 grid of 16x16 WMMA tiles, 4 v8f accumulators).
//
// Per K-step of 4: 2 global b64 (A fragments) + 2 ds b64 (B fragments)
// feeding 4 V_WMMA_F32_16X16X4_F32; all loop pointers bump by constants.
//
// Per ISA f32 layouts:
//   A 16x4 : lane = m + 16*(k>=2), vgpr = k%2
//   B 4x16 : lane = n + 16*(k>=2), vgpr = k%2
//   C 16x16: lane = n + 16*(m>=8), vgpr = m%8
// Tail-wave exit (tm >= tilesM) happens after the barrier and is
// wave-uniform, so EXEC is all-ones for every executed WMMA.
// ---------------------------------------------------------------------------
__global__ __launch_bounds__(256) void k_gemm_wmma(
        const float* __restrict__ A, const float* __restrict__ B,
        const float* __restrict__ bias, float* __restrict__ C,
        int M, int K, int N, int do_relu) {
    __shared__ float Bs[8192];                 // 32 KB: K x 32 panel, K <= 256

    const int lane = threadIdx.x & 31;
    const int wave = threadIdx.x >> 5;
    const int tilesM = M >> 5;                 // 32-row tiles
    const int tn = blockIdx.y;                 // shared by whole block

    // ---- cooperative stage of B[K][tn*32 .. tn*32+31] into LDS ----------
    // layout: Bs[(k>>1)*64 + j*2 + (k&1)]   (j = column within panel)
    for (int i = threadIdx.x; i < K * 32; i += 256) {
        const int k = i >> 5;
        const int j = i & 31;
        Bs[((k >> 1) << 6) + (j << 1) + (k & 1)] = B[(size_t)k * N + tn * 32 + j];
    }
    __syncthreads();

    const int tm = blockIdx.x * 8 + wave;
    if (tm >= tilesM) return;                  // wave-uniform exit

    const int half = lane >> 4;                // 0: k%4 in {0,1}, 1: {2,3}
    const int lid  = lane & 15;

    const float* A0 = A + (size_t)(tm * 32 + lid) * K + half * 2;
    const float* A1 = A0 + (size_t)16 * K;
    const float* Bs0 = &Bs[half * 64 + lid * 2];   // column lid
    const float* Bs1 = Bs0 + 32;                   // column lid+16

    v8f acc00 = {}, acc01 = {}, acc10 = {}, acc11 = {};
    for (int k = 0; k < K; k += 4) {
        v2f a0, a1;
        a0.x = A0[0];  a0.y = A0[1];
        a1.x = A1[0];  a1.y = A1[1];
        const v2f b0 = *(const v2f*)Bs0;           // ds_load_b64
        const v2f b1 = *(const v2f*)Bs1;           // ds_load_b64
        A0 += 4;  A1 += 4;  Bs0 += 128;  Bs1 += 128;
        acc00 = __builtin_amdgcn_wmma_f32_16x16x4_f32(
                    false, a0, false, b0, (short)0, acc00, false, false);
        acc01 = __builtin_amdgcn_wmma_f32_16x16x4_f32(
                    false, a0, false, b1, (short)0, acc01, false, false);
        acc10 = __builtin_amdgcn_wmma_f32_16x16x4_f32(
                    false, a1, false, b0, (short)0, acc10, false, false);
        acc11 = __builtin_amdgcn_wmma_f32_16x16x4_f32(
                    false, a1, false, b1, (short)0, acc11, false, false);
    }

    const float bv0 = bias ? bias[tn * 32 + lid]      : 0.0f;
    const float bv1 = bias ? bias[tn * 32 + 16 + lid] : 0.0f;
    float* Cb = C + (size_t)(tm * 32) * N + tn * 32 + lid;
    #pragma unroll
    for (int v = 0; v < 8; ++v) {
        const int mr = v + 8 * half;
        float x00 = acc00[v] + bv0;
        float x01 = acc01[v] + bv1;
        float x10 = acc10[v] + bv0;
        float x11 = acc11[v] + bv1;
        if (do_relu) {
            x00 = fmaxf(x00, 0.f); x01 = fmaxf(x01, 0.f);
            x10 = fmaxf(x10, 0.f); x11 = fmaxf(x11, 0.f);
        }
        Cb[(size_t)mr * N]             = x00;
        Cb[(size_t)mr * N + 16]        = x01;
        Cb[(size_t)(mr + 16) * N]      = x10;
        Cb[(size_t)(mr + 16) * N + 16] = x11;
    }
}

// ---------------------------------------------------------------------------
// Edge scatter:  acc[dst] += h[src]   (64 threads/edge, float4 per thread)
// Gather source stays L2-resident (102 MB < 192 MB L2).
// ---------------------------------------------------------------------------
__global__ __launch_bounds__(256) void k_scatter(
        const float* __restrict__ h, const int* __restrict__ src,
        const int* __restrict__ dst, float* __restrict__ acc, int n_edges) {
    long t = (long)blockIdx.x * blockDim.x + threadIdx.x;
    int e = (int)(t >> 6);
    if (e >= n_edges) return;
    int c = (int)(t & 63) * 4;
    int s = src[e], d = dst[e];
    const float4 v = *(const float4*)(h + (size_t)s * HID + c);
    float* a = acc + (size_t)d * HID + c;
    atomicAdd(a + 0, v.x);
    atomicAdd(a + 1, v.y);
    atomicAdd(a + 2, v.z);
    atomicAdd(a + 3, v.w);
}

// ---------------------------------------------------------------------------
// acc = relu?(acc + h)   (self-loop term of A_hat = A + I, fused activation)
// ---------------------------------------------------------------------------
__global__ __launch_bounds__(256) void k_add_act(
        float4* __restrict__ acc, const float4* __restrict__ h,
        long n4, int do_relu) {
    long i = (long)blockIdx.x * blockDim.x + threadIdx.x;
    if (i >= n4) return;
    float4 a = acc[i];
    float4 b = h[i];
    a.x += b.x; a.y += b.y; a.z += b.z; a.w += b.w;
    if (do_relu) {
        a.x = fmaxf(a.x, 0.f); a.y = fmaxf(a.y, 0.f);
        a.z = fmaxf(a.z, 0.f); a.w = fmaxf(a.w, 0.f);
    }
    acc[i] = a;
}

// ---------------------------------------------------------------------------
// Graph readout: pooled[idx[node]] += (scatter[node] + self[node])
// ---------------------------------------------------------------------------
__global__ __launch_bounds__(256) void k_pool(
        const float* __restrict__ acc, const float* __restrict__ y,
        const int* __restrict__ idx, float* __restrict__ pooled, int n_nodes) {
    long t = (long)blockIdx.x * blockDim.x + threadIdx.x;
    int node = (int)(t >> 6);
    if (node >= n_nodes) return;
    int c = (int)(t & 63) * 4;
    int g = idx[node];
    const float4 a = *(const float4*)(acc + (size_t)node * HID + c);
    const float4 b = *(const float4*)(y  + (size_t)node * HID + c);
    float* p = pooled + (size_t)g * HID + c;
    atomicAdd(p + 0, a.x + b.x);
    atomicAdd(p + 1, a.y + b.y);
    atomicAdd(p + 2, a.z + b.z);
    atomicAdd(p + 3, a.w + b.w);
}

// ---------------------------------------------------------------------------
// BatchNorm1d over [NGRAPH, HID], biased batch stats, training mode.
// One block, one thread per feature column.
// ---------------------------------------------------------------------------
__global__ __launch_bounds__(256) void k_bn(
        const float* __restrict__ pooled, const float* __restrict__ gamma,
        const float* __restrict__ beta, float* __restrict__ normed) {
    int c = threadIdx.x;
    float s = 0.f;
    for (int r = 0; r < NGRAPH; ++r) s += pooled[r * HID + c];
    float mean = s * (1.0f / NGRAPH);
    float var = 0.f;
    for (int r = 0; r < NGRAPH; ++r) {
        float d = pooled[r * HID + c] - mean;
        var += d * d;
    }
    var *= (1.0f / NGRAPH);
    float scale = rsqrtf(var + BN_EPS) * gamma[c];
    float shift = beta[c];
    for (int r = 0; r < NGRAPH; ++r)
        normed[r * HID + c] = (pooled[r * HID + c] - mean) * scale + shift;
}

// ---------------------------------------------------------------------------
// Head: logits = x @ W4 + b4  then log_softmax.  One thread per graph row
// (512 x 256 x 10 = 2.6 MFLOP — negligible).
// ---------------------------------------------------------------------------
__global__ __launch_bounds__(256) void k_head(
        const float* __restrict__ x, const float* __restrict__ W4,
        const float* __restrict__ b4, float* __restrict__ out) {
    int row = blockIdx.x * blockDim.x + threadIdx.x;
    if (row >= NGRAPH) return;
    float lg[NCLASS];
    #pragma unroll
    for (int c = 0; c < NCLASS; ++c) lg[c] = b4[c];
    for (int k = 0; k < HID; ++k) {
        float xv = x[(size_t)row * HID + k];
        #pragma unroll
        for (int c = 0; c < NCLASS; ++c) lg[c] += xv * W4[k * NCLASS + c];
    }
    float mx = lg[0];
    #pragma unroll
    for (int c = 1; c < NCLASS; ++c) mx = fmaxf(mx, lg[c]);
    float se = 0.f;
    #pragma unroll
    for (int c = 0; c < NCLASS; ++c) se += __expf(lg[c] - mx);
    float lse = __logf(se);
    #pragma unroll
    for (int c = 0; c < NCLASS; ++c)
        out[(size_t)row * NCLASS + c] = lg[c] - mx - lse;
}

// ---------------------------------------------------------------------------
// Driver
// ---------------------------------------------------------------------------
extern "C" void kernel_launch(void* const* d_in, const int* in_sizes, int n_in,
                              void* d_out, int out_size, void* d_ws, size_t ws_size,
                              hipStream_t stream) {
    const float* x_in  = (const float*)d_in[0];
    const int*   esrc  = (const int*)  d_in[1];
    const int*   edst  = (const int*)  d_in[2];
    const int*   idx   = (const int*)  d_in[3];
    const float* W1    = (const float*)d_in[4];
    const float* b1    = (const float*)d_in[5];
    const float* W2    = (const float*)d_in[6];
    const float* b2    = (const float*)d_in[7];
    const float* W3    = (const float*)d_in[8];
    const float* b3    = (const float*)d_in[9];
    const float* W4    = (const float*)d_in[10];
    const float* b4    = (const float*)d_in[11];
    const float* gamma = (const float*)d_in[12];
    const float* beta  = (const float*)d_in[13];

    const int n_nodes = in_sizes[0] / IN_DIM;   // 100000 (multiple of 32)
    const int n_edges = in_sizes[1];            // 3200000

    const size_t NH = (size_t)n_nodes * HID;    // 25.6M floats
    float* bufA   = (float*)d_ws;               // [N, HID] gemm out / gather src
    float* bufB   = bufA + NH;                  // [N, HID] scatter accumulator
    float* pooled = bufB + NH;                  // [G, HID]
    float* normed = pooled + (size_t)NGRAPH * HID;
    float* out3   = normed + (size_t)NGRAPH * HID;

    const long nh4 = (long)(NH / 4);
    const int  blkNH4 = (int)((nh4 + 255) / 256);
    const long scT = (long)n_edges * 64;
    const int  blkSc = (int)((scT + 255) / 256);
    const long poT = (long)n_nodes * 64;
    const int  blkPo = (int)((poT + 255) / 256);

    const dim3 gemm_grid_nodes((n_nodes / 32 + 7) / 8, HID / 32);  // (391, 8)
    const dim3 gemm_grid_graphs((NGRAPH / 32 + 7) / 8, HID / 32);  // (2, 8)

    // Layer 1: Y1 = x_in @ W1 + b1
    k_gemm_wmma<<<gemm_grid_nodes, 256, 0, stream>>>(
        x_in, W1, b1, bufA, n_nodes, IN_DIM, HID, 0);

    // spmm + relu: h1 = relu(scatter(Y1) + Y1)  -> bufB
    k_zero<<<blkNH4, 256, 0, stream>>>((float4*)bufB, nh4);
    k_scatter<<<blkSc, 256, 0, stream>>>(bufA, esrc, edst, bufB, n_edges);
    k_add_act<<<blkNH4, 256, 0, stream>>>((float4*)bufB, (const float4*)bufA, nh4, 1);

    // Layer 2: Y2 = h1 @ W2 + b2  -> bufA
    k_gemm_wmma<<<gemm_grid_nodes, 256, 0, stream>>>(
        bufB, W2, b2, bufA, n_nodes, HID, HID, 0);

    // spmm (no relu) fused into readout: pooled[g] += scatter(Y2) + Y2
    k_zero<<<blkNH4, 256, 0, stream>>>((float4*)bufB, nh4);
    k_scatter<<<blkSc, 256, 0, stream>>>(bufA, esrc, edst, bufB, n_edges);
    k_zero<<<(NGRAPH * HID / 4 + 255) / 256, 256, 0, stream>>>(
        (float4*)pooled, (long)(NGRAPH * HID / 4));
    k_pool<<<blkPo, 256, 0, stream>>>(bufB, bufA, idx, pooled, n_nodes);

    // BatchNorm
    k_bn<<<1, 256, 0, stream>>>(pooled, gamma, beta, normed);

    // Layer 3: out3 = relu(normed @ W3 + b3)
    k_gemm_wmma<<<gemm_grid_graphs, 256, 0, stream>>>(
        normed, W3, b3, out3, NGRAPH, HID, HID, 1);

    // Head: logits + log_softmax
    k_head<<<(NGRAPH + 255) / 256, 256, 0, stream>>>(out3, W4, b4, (float*)d_out);
}